// GraphEmbedding_798863917733
// MI455X (gfx1250) — compile-verified
//
#include <hip/hip_runtime.h>

// CDNA5 / gfx1250: wave32. f32 WMMA vector types.
typedef float v2f __attribute__((ext_vector_type(2)));
typedef float v8f __attribute__((ext_vector_type(8)));

#define NUM_EDGES     3200001
#define NUM_EDGES_PAD 3200004            // pad to 16B alignment for table 2
#define BATCH         32768
#define NUM_TARGETS   8
#define MAX_LENGTH    64
#define PAIRS         (BATCH * NUM_TARGETS)   // 262144
#define WS_NEEDED     ((size_t)2 * NUM_EDGES_PAD * sizeof(float))

// Precise softplus for the (cheap, per-edge) table precompute.
__device__ __forceinline__ float softplus_precise(float x) {
    return fmaxf(x, 0.0f) + log1pf(expf(-fabsf(x)));
}

// Fast softplus on native v_exp_f32 / v_log_f32 (TRANS pipe, co-executes with
// VALU): max(x,0) + ln2 * log2(1 + exp2(-|x| * log2e)). ~1e-7 abs error vs
// log1p form; softplus(-1e4) -> 0 exactly, so the dummy edge contributes 0.
__device__ __forceinline__ float softplus_fast(float x) {
    float t = __builtin_amdgcn_exp2f(-1.442695041f * fabsf(x));
    return fmaxf(x, 0.0f) + 0.6931471806f * __builtin_amdgcn_logf(1.0f + t);
}

// Per-edge tables: spw[i] = softplus(w[i]), lsa[i] = log_sigmoid(a[i]).
// Only 3.2M edges vs 16.7M path gathers -> 5x less transcendental work.
__global__ __launch_bounds__(256) void GraphEmbedding_tables_kernel(
    const float* __restrict__ adj, const float* __restrict__ wlg,
    float* __restrict__ spw, float* __restrict__ lsa)
{
    int i = blockIdx.x * blockDim.x + threadIdx.x;
    if (i < NUM_EDGES) {
        spw[i] = softplus_precise(wlg[i]);
        lsa[i] = -softplus_precise(-adj[i]);   // log_sigmoid(a) = -softplus(-a)
    }
}

// One wave32 handles 16 (batch,target) pairs.
//   lane L (<16) : pair n=L,    positions  0..31 -> partials b0, b1
//   lane L (>=16): pair n=L-16, positions 32..63 -> partials b0, b1
// This is exactly the V_WMMA_F32_16X16X4_F32 B-matrix layout (4x16, column n
// in lanes n and n+16, 2 VGPRs). A = all-ones row-0 selector (lanes 0,16) ->
// D[0][n] = weight sum of pair n (C VGPR0, lanes 0-15). Second WMMA with
// row-8 selector (lanes 8,24) and logp partials accumulates D[8][n] into the
// same C VGPR0 on lanes 16-31. Summing all 4 K-slots with coefficient 1 makes
// the result invariant to the K permutation across half-waves.
template <bool USE_TABLES>
__global__ __launch_bounds__(256) void GraphEmbedding_path_kernel(
    const float* __restrict__ adj,    // edge_adjacency_logits [NUM_EDGES]
    const float* __restrict__ wlg,    // edge_weight_logits    [NUM_EDGES]
    const float* __restrict__ defd,   // default_distance      [1]
    const int*   __restrict__ paths,  // target_paths          [PAIRS * 64]
    const float* __restrict__ spw,    // softplus table (or null)
    const float* __restrict__ lsa,    // log_sigmoid table (or null)
    float*       __restrict__ out)    // [3*PAIRS]: dist | logp | found
{
    const int lane = threadIdx.x & 31;
    const int wave = blockIdx.x * (blockDim.x >> 5) + (threadIdx.x >> 5);
    const int n    = lane & 15;            // pair column within wave's tile
    const int p    = (wave << 4) + n;      // this lane's pair index
    const int hoff = (lane >> 4) << 5;     // 0 or 32: which half of the path

    const int* __restrict__ pp = paths + (size_t)p * MAX_LENGTH + hoff;

    float b0w = 0.0f, b1w = 0.0f;   // weight-sum partials (two K-slots)
    float b0p = 0.0f, b1p = 0.0f;   // logp-sum partials
    int   nz  = 0;

#pragma unroll
    for (int j = 0; j < 16; ++j) {
        int idx = pp[j];
        nz |= idx;
        if (USE_TABLES) {
            b0w += spw[idx];
            b0p += lsa[idx];
        } else {
            b0w += softplus_fast(wlg[idx]);
            b0p -= softplus_fast(-adj[idx]);
        }
    }
#pragma unroll
    for (int j = 16; j < 32; ++j) {
        int idx = pp[j];
        nz |= idx;
        if (USE_TABLES) {
            b1w += spw[idx];
            b1p += lsa[idx];
        } else {
            b1w += softplus_fast(wlg[idx]);
            b1p -= softplus_fast(-adj[idx]);
        }
    }

    // found flag: bit L set iff lane L's 32 positions contain a nonzero edge
    unsigned mask = __builtin_amdgcn_ballot_w32(nz != 0);

    // A-matrix selectors: row 0 lives in lanes {0,16}, row 8 in lanes {8,24}
    float s0 = (n == 0) ? 1.0f : 0.0f;
    float s8 = (n == 8) ? 1.0f : 0.0f;
    v2f Aw = { s0, s0 };
    v2f Ap = { s8, s8 };
    v2f Bw = { b0w, b1w };
    v2f Bp = { b0p, b1p };
    v8f c  = { 0.f, 0.f, 0.f, 0.f, 0.f, 0.f, 0.f, 0.f };

    // D = A*B + C (exact f32 FMA, RTE). EXEC is all ones here (no divergence).
    c = __builtin_amdgcn_wmma_f32_16x16x4_f32(false, Aw, false, Bw,
                                              (short)0, c, false, false);
    c = __builtin_amdgcn_wmma_f32_16x16x4_f32(false, Ap, false, Bp,
                                              (short)0, c, false, false);

    // c[0]: lanes 0-15 = weight sum of pair n; lanes 16-31 = logp sum of pair n
    float s = c[0];
    bool found = (((mask >> n) | (mask >> (n + 16))) & 1u) != 0u;

    if (lane < 16) {
        out[p]             = found ? s : defd[0];        // target_distances
        out[2 * PAIRS + p] = found ? 1.0f : 0.0f;        // found_target
    } else {
        out[PAIRS + p]     = found ? s : 0.0f;           // logp_target_paths
    }
}

extern "C" void kernel_launch(void* const* d_in, const int* in_sizes, int n_in,
                              void* d_out, int out_size, void* d_ws, size_t ws_size,
                              hipStream_t stream) {
    const float* adj   = (const float*)d_in[0];  // edge_adjacency_logits
    const float* wlg   = (const float*)d_in[1];  // edge_weight_logits
    const float* defd  = (const float*)d_in[2];  // default_distance
    const int*   paths = (const int*)d_in[3];    // target_paths
    float*       out   = (float*)d_out;

    // 16 pairs/wave * 8 waves/block = 128 pairs/block -> 2048 blocks
    const int blocks = PAIRS / 128;

    if (ws_size >= WS_NEEDED) {
        float* spw = (float*)d_ws;
        float* lsa = spw + NUM_EDGES_PAD;
        GraphEmbedding_tables_kernel<<<(NUM_EDGES + 255) / 256, 256, 0, stream>>>(
            adj, wlg, spw, lsa);
        GraphEmbedding_path_kernel<true><<<blocks, 256, 0, stream>>>(
            adj, wlg, defd, paths, spw, lsa, out);
    } else {
        GraphEmbedding_path_kernel<false><<<blocks, 256, 0, stream>>>(
            adj, wlg, defd, paths, (const float*)nullptr, (const float*)nullptr, out);
    }
}